// CantorExpert_57174604644749
// MI455X (gfx1250) — compile-verified
//
#include <hip/hip_runtime.h>
#include <math.h>

typedef float v2f __attribute__((ext_vector_type(2)));
typedef float v8f __attribute__((ext_vector_type(8)));

#define FULL_DIM   4096
#define SLICE      512
#define BAND_OFF   1536          // EXPERT_ID(3) * SLICE
#define GATE_HID   128
#define EXPERT_DIM 1024
#define FP_MIN     0.375f
#define FP_MAX     0.5f

// ---------------------------------------------------------------- helpers
__device__ __forceinline__ uint32_t lds_off_of(const void* p) {
  // generic pointer into __shared__: low 32 bits == LDS byte offset (flat-LDS aperture)
  return (uint32_t)(uintptr_t)p;
}
__device__ __forceinline__ void async_g2l_b128(uint32_t ldsoff, const float* g) {
  asm volatile("global_load_async_to_lds_b128 %0, %1, off"
               :: "v"(ldsoff), "v"(g) : "memory");
}
__device__ __forceinline__ void wait_async_le0() {
  asm volatile("s_wait_asynccnt 0" ::: "memory");
}
__device__ __forceinline__ void wait_async_le6() {
  // async-load "done" returns in order -> waiting to <=6 completes the PREVIOUS
  // 6-instruction copy group while the next 6 stay in flight
  asm volatile("s_wait_asynccnt 6" ::: "memory");
}
__device__ __forceinline__ float sigmoidf_(float x) { return 1.0f / (1.0f + expf(-x)); }

// ---------------------------------------------------------------- 1) mask + compaction (single wave32)
__global__ __launch_bounds__(32) void mask_compact_kernel(const float* __restrict__ fps,
                                                          int* __restrict__ idx,
                                                          float* __restrict__ mask_out,
                                                          int P) {
  const int lane = threadIdx.x;
  int base = 0;
  for (int s = 0; s < P; s += 32) {
    const int i = s + lane;
    const float f = fps[i];
    const bool pred = (f >= FP_MIN) && (f < FP_MAX);
    mask_out[i] = pred ? 1.0f : 0.0f;
    const unsigned bal = (unsigned)__ballot(pred);          // wave32 -> low 32 bits
    if (pred) idx[base + __popc(bal & ((1u << lane) - 1u))] = i;
    base += __popc(bal);
  }
}

// ---------------------------------------------------------------- 2) gating MLP + feature scaling
__global__ __launch_bounds__(128) void gate_kernel(const float* __restrict__ tokens,
                                                   const int* __restrict__ idx,
                                                   const float* __restrict__ w1,
                                                   const float* __restrict__ b1,
                                                   const float* __restrict__ w2,
                                                   const float* __restrict__ b2,
                                                   const float* __restrict__ alpha,
                                                   float* __restrict__ F,
                                                   int p, int P) {
  __shared__ float f[SLICE];
  __shared__ float red[GATE_HID];
  const int t   = threadIdx.x;
  const int blk = blockIdx.x;               // 0 .. B*p-1
  const int b   = blk / p;
  const int j   = blk - b * p;
  const int patch = idx[j];
  const float* src = tokens + ((size_t)b * P + patch) * FULL_DIM + BAND_OFF;
  for (int i = t; i < SLICE; i += 128) f[i] = src[i];
  __syncthreads();
  // h = gelu(f @ w1.T + b1); one hidden unit per thread
  float h = b1[t];
  const float* w1r = w1 + (size_t)t * SLICE;
  #pragma unroll 8
  for (int i = 0; i < SLICE; ++i) h = fmaf(f[i], w1r[i], h);
  const float g = 0.5f * h * (1.0f + erff(h * 0.70710678118654752f)); // exact GELU
  red[t] = g * w2[t];
  __syncthreads();
  for (int s = 64; s > 0; s >>= 1) { if (t < s) red[t] += red[t + s]; __syncthreads(); }
  const float gate = sigmoidf_(red[0] + b2[0]);
  const float aw   = sigmoidf_(alpha[0]);
  const float scale = gate * aw + (1.0f - aw);
  float* dst = F + (size_t)blk * SLICE;
  for (int i = t; i < SLICE; i += 128) dst[i] = f[i] * scale;
}

// ---------------------------------------------------------------- 3) WMMA f32 GEMM: C[N,1024] = F[N,512] @ W[1024,512]^T
//     double-buffered async global->LDS pipeline (ASYNCcnt in-order completion)
#define TM  128
#define TN  64
#define KC  32
#define LDA 36   // padded pitch (floats): 144B rows -> 16B aligned, bank-conflict-free frags
#define LDB 36

__global__ __launch_bounds__(256) void qkv_wmma_kernel(const float* __restrict__ F,
                                                       const float* __restrict__ W,
                                                       float* __restrict__ C,
                                                       int N) {
  __shared__ __align__(16) float sA[2][TM * LDA];
  __shared__ __align__(16) float sB[2][TN * LDB];
  const int tid  = threadIdx.x;
  const int lane = tid & 31;
  const int wave = tid >> 5;                 // 0..7, each owns a 16-row strip
  const int half = lane >> 4;                // 0 | 1 (K halves of the f32 frag)
  const int l16  = lane & 15;
  const int mBlock = blockIdx.x * TM;
  const int nBlock = blockIdx.y * TN;

  v8f acc[4];
  #pragma unroll
  for (int a = 0; a < 4; ++a) acc[a] = (v8f)0.0f;

  // per-thread copy-slice coordinates (6 async b128 per thread per tile)
  const int aRowIdx[4] = { (tid          ) >> 3, (tid + 256) >> 3,
                           (tid + 512    ) >> 3, (tid + 768) >> 3 };
  const int aC4       = (tid & 7) << 2;
  const int bRowIdx[2] = { tid >> 3, (tid + 256) >> 3 };

  uint32_t sAoff[2] = { lds_off_of(sA[0]), lds_off_of(sA[1]) };
  uint32_t sBoff[2] = { lds_off_of(sB[0]), lds_off_of(sB[1]) };

  // ---- issue one K-tile's 6 async copies into buffer `bf`
  auto issue_tile = [&](int k0, int bf) {
    #pragma unroll
    for (int s = 0; s < 4; ++s) {
      const int row = aRowIdx[s];
      int grow = mBlock + row; if (grow > N - 1) grow = N - 1;   // clamp (stores guarded)
      async_g2l_b128(sAoff[bf] + (uint32_t)(row * LDA + aC4) * 4u,
                     F + (size_t)grow * SLICE + k0 + aC4);
    }
    #pragma unroll
    for (int s = 0; s < 2; ++s) {
      const int n = bRowIdx[s];
      async_g2l_b128(sBoff[bf] + (uint32_t)(n * LDB + aC4) * 4u,
                     W + (size_t)(nBlock + n) * SLICE + k0 + aC4);
    }
  };

  issue_tile(0, 0);                     // prologue: stage first tile
  int buf = 0;
  for (int k0 = 0; k0 < SLICE; k0 += KC) {
    __syncthreads();                    // all waves done reading buf^1 (prev iter)
    const bool has_next = (k0 + KC) < SLICE;
    if (has_next) issue_tile(k0 + KC, buf ^ 1);   // prefetch next tile
    if (has_next) wait_async_le6(); else wait_async_le0();  // current tile landed
    __syncthreads();                    // publish current buffer to all waves

    const float* aRow = sA[buf] + (wave * 16 + l16) * LDA;
    const float* bBase = sB[buf];
    #pragma unroll
    for (int kk = 0; kk < KC; kk += 4) {
      // A frag (16x4 f32): lanes 0-15 -> K=kk,kk+1 ; lanes 16-31 -> K=kk+2,kk+3
      const v2f af = *(const v2f*)(aRow + kk + 2 * half);
      #pragma unroll
      for (int a = 0; a < 4; ++a) {
        // B frag (4x16 f32): B[k,n] = W[n,k] = sB[n*LDB + k]
        const v2f bf2 = *(const v2f*)(bBase + (a * 16 + l16) * LDB + kk + 2 * half);
        acc[a] = __builtin_amdgcn_wmma_f32_16x16x4_f32(
            false, af, false, bf2, (short)0, acc[a], false, false);
      }
    }
    buf ^= 1;
  }

  // C/D layout: VGPR r: lanes 0-15 -> M=r, lanes 16-31 -> M=r+8
  #pragma unroll
  for (int a = 0; a < 4; ++a) {
    const int col = nBlock + a * 16 + l16;
    #pragma unroll
    for (int r = 0; r < 8; ++r) {
      const int row = mBlock + wave * 16 + r + 8 * half;
      if (row < N) C[(size_t)row * EXPERT_DIM + col] = acc[a][r];
    }
  }
}

// ---------------------------------------------------------------- 4) normalize pentachoron rows
__global__ __launch_bounds__(256) void dirs_kernel(const float* __restrict__ penta,
                                                   float* __restrict__ dirs) {
  __shared__ float red[256];
  const int v = blockIdx.x, t = threadIdx.x;
  const float* row = penta + (size_t)v * EXPERT_DIM;
  float s = 0.0f;
  for (int i = t; i < EXPERT_DIM; i += 256) { const float x = row[i]; s = fmaf(x, x, s); }
  red[t] = s; __syncthreads();
  for (int st = 128; st > 0; st >>= 1) { if (t < st) red[t] += red[t + st]; __syncthreads(); }
  const float inv = 1.0f / sqrtf(red[0]);
  for (int i = t; i < EXPERT_DIM; i += 256) dirs[(size_t)v * EXPERT_DIM + i] = row[i] * inv;
}

// ---------------------------------------------------------------- 5) affinities: [5,N] dots for Q and K
__global__ __launch_bounds__(128) void aff_kernel(const float* __restrict__ Q,
                                                  const float* __restrict__ K,
                                                  const float* __restrict__ dirs,
                                                  float* __restrict__ Qaff,
                                                  float* __restrict__ Kaff,
                                                  int N) {
  __shared__ float red[10 * 128];
  const int tok = blockIdx.x, t = threadIdx.x;
  const float* q = Q + (size_t)tok * EXPERT_DIM;
  const float* k = K + (size_t)tok * EXPERT_DIM;
  float aq[5] = {0, 0, 0, 0, 0}, ak[5] = {0, 0, 0, 0, 0};
  for (int d = t; d < EXPERT_DIM; d += 128) {
    const float qv = q[d], kv = k[d];
    #pragma unroll
    for (int v = 0; v < 5; ++v) {
      const float dv = dirs[(size_t)v * EXPERT_DIM + d];
      aq[v] = fmaf(qv, dv, aq[v]);
      ak[v] = fmaf(kv, dv, ak[v]);
    }
  }
  #pragma unroll
  for (int v = 0; v < 5; ++v) { red[v * 128 + t] = aq[v]; red[(5 + v) * 128 + t] = ak[v]; }
  __syncthreads();
  for (int s = 64; s > 0; s >>= 1) {
    if (t < s) {
      #pragma unroll
      for (int v = 0; v < 10; ++v) red[v * 128 + t] += red[v * 128 + t + s];
    }
    __syncthreads();
  }
  if (t < 5)            Qaff[(size_t)t * N + tok]       = red[t * 128];
  else if (t < 10)      Kaff[(size_t)(t - 5) * N + tok] = red[t * 128];
}

// ---------------------------------------------------------------- launcher
extern "C" void kernel_launch(void* const* d_in, const int* in_sizes, int n_in,
                              void* d_out, int out_size, void* d_ws, size_t ws_size,
                              hipStream_t stream) {
  const float* tokens = (const float*)d_in[0];
  const float* fps    = (const float*)d_in[1];
  const float* alpha  = (const float*)d_in[2];
  const float* gw1    = (const float*)d_in[3];
  const float* gb1    = (const float*)d_in[4];
  const float* gw2    = (const float*)d_in[5];
  const float* gb2    = (const float*)d_in[6];
  const float* wq     = (const float*)d_in[7];
  const float* wk     = (const float*)d_in[8];
  const float* wv     = (const float*)d_in[9];
  const float* penta  = (const float*)d_in[10];

  const int P = in_sizes[1];                               // 4096
  const int B = in_sizes[0] / (P * FULL_DIM);              // 4
  // out_size = 3*B*p*1024 + 2*5*B*p + P  ->  recover p
  const int p = (out_size - P) / (B * (3 * EXPERT_DIM + 10));
  const int N = B * p;

  float* outQ    = (float*)d_out;
  float* outK    = outQ + (size_t)N * EXPERT_DIM;
  float* outV    = outK + (size_t)N * EXPERT_DIM;
  float* outKaff = outV + (size_t)N * EXPERT_DIM;
  float* outQaff = outKaff + (size_t)5 * N;
  float* outMask = outQaff + (size_t)5 * N;

  int*   idx  = (int*)d_ws;
  float* dirs = (float*)((char*)d_ws + (((size_t)P * 4 + 255) & ~(size_t)255));
  float* F    = dirs + (size_t)5 * EXPERT_DIM;

  mask_compact_kernel<<<1, 32, 0, stream>>>(fps, idx, outMask, P);
  dirs_kernel<<<5, 256, 0, stream>>>(penta, dirs);

  if (p > 0) {
    gate_kernel<<<N, 128, 0, stream>>>(tokens, idx, gw1, gb1, gw2, gb2, alpha, F, p, P);
    dim3 grid((N + TM - 1) / TM, EXPERT_DIM / TN, 1);
    qkv_wmma_kernel<<<grid, 256, 0, stream>>>(F, wq, outQ, N);
    qkv_wmma_kernel<<<grid, 256, 0, stream>>>(F, wk, outK, N);
    qkv_wmma_kernel<<<grid, 256, 0, stream>>>(F, wv, outV, N);
    aff_kernel<<<N, 128, 0, stream>>>(outQ, outK, dirs, outQaff, outKaff, N);
  }
}